// MosaicSDF_27599459845006
// MI455X (gfx1250) — compile-verified
//
#include <hip/hip_runtime.h>
#include <hip/hip_bf16.h>

#define N_POINTS 256
#define N_GRIDS  1024
#define KDIM     7
#define KCELLS   343
#define EPSV     1e-4f

typedef __attribute__((ext_vector_type(2))) float v2f;
typedef __attribute__((ext_vector_type(8))) float v8f;

#if __has_builtin(__builtin_amdgcn_wmma_f32_16x16x4_f32)
#define HAVE_WMMA4 1
#else
#define HAVE_WMMA4 0
#endif

// Per-(point,grid) 7x7x7 interpolation. rel = local coords in grid frame.
// span_k = -1 + k/3, coord_step = 1/3, so (rel - span_k)*3 == (3*rel + 3) - k.
__device__ __forceinline__ float interp_eval(const float* __restrict__ sdfv,
                                             float rx, float ry, float rz) {
    const float bx = fmaf(rx, 3.0f, 3.0f);
    const float by = fmaf(ry, 3.0f, 3.0f);
    const float bz = fmaf(rz, 3.0f, 3.0f);
    float wsum = 0.0f, wv = 0.0f;
    int idx = 0;
    for (int i = 0; i < KDIM; ++i) {
        const float dx = bx - (float)i;
        const float sx = dx * dx;
        for (int j = 0; j < KDIM; ++j) {
            const float dy = by - (float)j;
            const float sxy = fmaf(dy, dy, sx);
#pragma unroll
            for (int l = 0; l < KDIM; ++l) {
                const float dz = bz - (float)l;
                const float d2 = fmaf(dz, dz, sxy);
                const float dist = __builtin_amdgcn_sqrtf(d2);
                const float w = (dist <= 1.0f) ? (1.0f - dist) : 0.0f;
                wsum += w;
                wv = fmaf(w, sdfv[idx + l], wv);
            }
            idx += KDIM;
        }
    }
    return (wsum > EPSV) ? (wv / wsum) : 0.0f;
}

// One block per grid g (1024 blocks), one thread per point (256 threads).
// Stages the 343 SDF values in LDS; all lanes broadcast-read the same cell.
__global__ void __launch_bounds__(256)
mosaic_pair_kernel(const float* __restrict__ pts, const float* __restrict__ ctr,
                   const float* __restrict__ scl, const float* __restrict__ sdf,
                   float* __restrict__ num, float* __restrict__ den) {
    const int g = blockIdx.x;
    const int n = threadIdx.x;

    __shared__ float s_sdf[KCELLS];
    for (int i = threadIdx.x; i < KCELLS; i += 256)
        s_sdf[i] = sdf[g * KCELLS + i];
    __syncthreads();

    const float s  = scl[g];
    const float cx = ctr[g * 3 + 0], cy = ctr[g * 3 + 1], cz = ctr[g * 3 + 2];
    const float rx = (pts[n * 3 + 0] - cx) / s;
    const float ry = (pts[n * 3 + 1] - cy) / s;
    const float rz = (pts[n * 3 + 2] - cz) / s;

    const float gdist = __builtin_amdgcn_sqrtf(fmaf(rx, rx, fmaf(ry, ry, rz * rz)));
    const float gw = fmaxf(1.0f + EPSV - gdist, 0.0f);

    float interp = 0.0f;
    if (gw > 0.0f)  // ~99% of pairs skip the 343-cell loop
        interp = interp_eval(s_sdf, rx, ry, rz);

    num[g * N_POINTS + n] = interp * gw;   // coalesced
    den[g * N_POINTS + n] = gw;
}

#if HAVE_WMMA4
// Row-sum of num/den (1024 grids per point) as GEMM vs ones using
// V_WMMA_F32_16X16X4_F32. One wave (32 threads) per 16-point group.
// A layout (16x4 f32): lanes 0-15 -> K=0,1 in a.x/a.y; lanes 16-31 -> K=2,3.
__global__ void __launch_bounds__(32)
mosaic_reduce_wmma(const float* __restrict__ num, const float* __restrict__ den,
                   float* __restrict__ out) {
    const int lane = threadIdx.x;          // 0..31
    const int m0   = blockIdx.x * 16;      // 16 blocks * 16 points
    const int m    = m0 + (lane & 15);
    const int koff = (lane >> 4) * 2;

    v2f ones; ones.x = 1.0f; ones.y = 1.0f;   // B = all ones -> layout-free
    v8f cN = {}; v8f cD = {};

    for (int k0 = 0; k0 < N_GRIDS; k0 += 4) {
        const int r0 = k0 + koff;
        v2f aN, aD;
        aN.x = num[(r0 + 0) * N_POINTS + m];
        aN.y = num[(r0 + 1) * N_POINTS + m];
        aD.x = den[(r0 + 0) * N_POINTS + m];
        aD.y = den[(r0 + 1) * N_POINTS + m];
        cN = __builtin_amdgcn_wmma_f32_16x16x4_f32(false, aN, false, ones,
                                                   (short)0, cN, false, false);
        cD = __builtin_amdgcn_wmma_f32_16x16x4_f32(false, aD, false, ones,
                                                   (short)0, cD, false, false);
    }

    // D layout: VGPR v holds M=v (lanes 0-15, N=lane) and M=v+8 (lanes 16-31).
    // Lanes 0 and 16 own column N=0.
    if ((lane & 15) == 0) {
        const int base = m0 + (lane >> 4) * 8;
#pragma unroll
        for (int v = 0; v < 8; ++v) {
            const float ns = cN[v];
            const float ds = cD[v];
            float r = (ds < EPSV) ? 1.0f : ((ds > EPSV) ? ns / ds : 0.0f);
            out[base + v] = r;
        }
    }
}
#endif

// VALU reduction fallback (used only if the f32 WMMA builtin is absent).
__global__ void __launch_bounds__(256)
mosaic_reduce_plain(const float* __restrict__ num, const float* __restrict__ den,
                    float* __restrict__ out) {
    const int n = blockIdx.x;
    const int t = threadIdx.x;
    float aN = 0.0f, aD = 0.0f;
    for (int g = t; g < N_GRIDS; g += 256) {
        aN += num[g * N_POINTS + n];
        aD += den[g * N_POINTS + n];
    }
    __shared__ float sN[256], sD[256];
    sN[t] = aN; sD[t] = aD;
    __syncthreads();
    for (int off = 128; off > 0; off >>= 1) {
        if (t < off) { sN[t] += sN[t + off]; sD[t] += sD[t + off]; }
        __syncthreads();
    }
    if (t == 0) {
        const float ds = sD[0], ns = sN[0];
        out[n] = (ds < EPSV) ? 1.0f : ((ds > EPSV) ? ns / ds : 0.0f);
    }
}

// Single-kernel fallback when workspace is too small: one block per point.
__global__ void __launch_bounds__(256)
mosaic_mono_kernel(const float* __restrict__ pts, const float* __restrict__ ctr,
                   const float* __restrict__ scl, const float* __restrict__ sdf,
                   float* __restrict__ out) {
    const int n = blockIdx.x;
    const int t = threadIdx.x;
    const float px = pts[n * 3 + 0], py = pts[n * 3 + 1], pz = pts[n * 3 + 2];

    float aN = 0.0f, aD = 0.0f;
    for (int g = t; g < N_GRIDS; g += 256) {
        const float s  = scl[g];
        const float rx = (px - ctr[g * 3 + 0]) / s;
        const float ry = (py - ctr[g * 3 + 1]) / s;
        const float rz = (pz - ctr[g * 3 + 2]) / s;
        const float gdist = __builtin_amdgcn_sqrtf(fmaf(rx, rx, fmaf(ry, ry, rz * rz)));
        const float gw = fmaxf(1.0f + EPSV - gdist, 0.0f);
        if (gw > 0.0f) {
            const float interp = interp_eval(&sdf[g * KCELLS], rx, ry, rz);
            aN = fmaf(interp, gw, aN);
        }
        aD += gw;
    }
    __shared__ float sN[256], sD[256];
    sN[t] = aN; sD[t] = aD;
    __syncthreads();
    for (int off = 128; off > 0; off >>= 1) {
        if (t < off) { sN[t] += sN[t + off]; sD[t] += sD[t + off]; }
        __syncthreads();
    }
    if (t == 0) {
        const float ds = sD[0], ns = sN[0];
        out[n] = (ds < EPSV) ? 1.0f : ((ds > EPSV) ? ns / ds : 0.0f);
    }
}

extern "C" void kernel_launch(void* const* d_in, const int* in_sizes, int n_in,
                              void* d_out, int out_size, void* d_ws, size_t ws_size,
                              hipStream_t stream) {
    (void)in_sizes; (void)n_in; (void)out_size;
    const float* pts = (const float*)d_in[0];   // (256,3)
    const float* ctr = (const float*)d_in[1];   // (1024,3)
    const float* scl = (const float*)d_in[2];   // (1024,)
    const float* sdf = (const float*)d_in[3];   // (1024,7,7,7)
    float* out = (float*)d_out;                 // (256,)

    const size_t need = (size_t)2 * N_GRIDS * N_POINTS * sizeof(float);
    if (ws_size >= need) {
        float* num = (float*)d_ws;
        float* den = num + (size_t)N_GRIDS * N_POINTS;
        mosaic_pair_kernel<<<dim3(N_GRIDS), dim3(256), 0, stream>>>(
            pts, ctr, scl, sdf, num, den);
#if HAVE_WMMA4
        mosaic_reduce_wmma<<<dim3(N_POINTS / 16), dim3(32), 0, stream>>>(num, den, out);
#else
        mosaic_reduce_plain<<<dim3(N_POINTS), dim3(256), 0, stream>>>(num, den, out);
#endif
    } else {
        mosaic_mono_kernel<<<dim3(N_POINTS), dim3(256), 0, stream>>>(
            pts, ctr, scl, sdf, out);
    }
}